// ObjectDetectionPostProcessor_72224170049914
// MI455X (gfx1250) — compile-verified
//
#include <hip/hip_runtime.h>
#include <stdint.h>

#define NUM_CLASSES 80
#define NPB   8400        // anchors per batch (6400 + 1600 + 400)
#define NSORT 16384       // next pow2 for bitonic sort
#define BATCH 32
#define THRESH 0.3f

typedef uint32_t v4u __attribute__((ext_vector_type(4)));
typedef int      v8i __attribute__((ext_vector_type(8)));
typedef int      v4i __attribute__((ext_vector_type(4)));

#if defined(__has_builtin)
#if __has_builtin(__builtin_amdgcn_tensor_load_to_lds)
#define HAVE_TDM 1
#endif
#endif

__device__ __forceinline__ float sigmoidf(float x) {
    return 1.0f / (1.0f + expf(-x));
}

#ifdef HAVE_TDM
// Issue one TDM DMA of `ndw` contiguous dwords (ndw <= 65535) global -> LDS.
// Tracked on the issuing wave's TENSORcnt.
__device__ __forceinline__ void tdm_copy_dwords(uint32_t lds_addr,
                                                const void* gptr,
                                                uint32_t ndw) {
    uint64_t gaddr = (uint64_t)(uintptr_t)gptr;
    // D# group0: count=1, lds_addr, global_addr[56:0], type=2 ("image")
    v4u g0 = { 1u, lds_addr, (uint32_t)gaddr,
               (uint32_t)((gaddr >> 32) & 0x01FFFFFFu) | 0x80000000u };
    // D# group1: data_size=2 (4B); tensor_dim0 = tile_dim0 = ndw;
    //            tensor_dim1 = tile_dim1 = 1; strides = ndw
    v8i g1 = { (int)0x20000u,                          // mask=0, data_size=4B
               (int)((ndw & 0xFFFFu) << 16),           // tensor_dim0[15:0]
               (int)(((ndw >> 16) & 0xFFFFu) | (1u << 16)), // dim0 hi | dim1=1
               (int)((ndw & 0xFFFFu) << 16),           // tile_dim0
               1,                                      // tile_dim1=1, tile_dim2=0
               (int)ndw,                               // stride0[31:0]
               (int)((ndw & 0xFFFFu) << 16),           // stride0 hi | stride1 lo
               (int)(ndw >> 16) };                     // stride1[47:16]
    v4i gz = { 0, 0, 0, 0 };
#if __has_include(<hip/amd_detail/amd_gfx1250_TDM.h>)
    v8i gz8 = { 0, 0, 0, 0, 0, 0, 0, 0 };              // clang-23 6-arg form
    __builtin_amdgcn_tensor_load_to_lds(g0, g1, gz, gz, gz8, 0);
#else
    __builtin_amdgcn_tensor_load_to_lds(g0, g1, gz, gz, 0);  // ROCm 7.2 5-arg
#endif
}
#endif

// ---------------------------------------------------------------------------
// Kernel 1: fused decode.  One thread per anchor.  Reads 85 channels
// (coalesced per channel), computes boxes / masked max score / argmax class.
// ---------------------------------------------------------------------------
__global__ void decode_kernel(const float* __restrict__ out0,
                              const float* __restrict__ out1,
                              const float* __restrict__ out2,
                              float* __restrict__ boxes,   // [B*NPB*4]
                              float* __restrict__ scores,  // [B*NPB] masked
                              int*   __restrict__ clsid)   // [B*NPB]
{
    int gid = blockIdx.x * blockDim.x + threadIdx.x;
    if (gid >= BATCH * NPB) return;
    int b = gid / NPB;
    int a = gid - b * NPB;

    const float* src; int H, W, off; float stride;
    if (a < 6400)      { src = out0; H = 80; W = 80; off = 0;    stride = 8.f;  }
    else if (a < 8000) { src = out1; H = 40; W = 40; off = 6400; stride = 16.f; }
    else               { src = out2; H = 20; W = 20; off = 8000; stride = 32.f; }
    int al = a - off;
    int y  = al / W, x = al - y * W;
    int HW = H * W;
    const float* p = src + (size_t)b * 85 * HW + al;   // channel stride = HW

    float tx = p[0 * HW], ty = p[1 * HW];
    float tw = p[2 * HW], th = p[3 * HW];
    float tobj = p[4 * HW];

    __builtin_prefetch(p + 5 * HW, 0, 1);              // global_prefetch_b8
    float bestl = -INFINITY; int bestc = 0;
    #pragma unroll 4
    for (int c = 0; c < NUM_CLASSES; ++c) {
        float v = p[(5 + c) * HW];
        if (v > bestl) { bestl = v; bestc = c; }       // first max, like argmax
    }

    float cx = (tx + (float)x) * stride;
    float cy = (ty + (float)y) * stride;
    float bw = expf(tw) * stride;
    float bh = expf(th) * stride;
    float obj = sigmoidf(tobj);
    float sc  = sigmoidf(bestl) * obj;                 // = max over classes
    float masked = (sc > THRESH) ? sc : -1.0f;

    size_t o = (size_t)b * NPB + a;
    float4 bx = make_float4(cx - 0.5f * bw, cy - 0.5f * bh,
                            cx + 0.5f * bw, cy + 0.5f * bh);
    reinterpret_cast<float4*>(boxes)[o] = bx;
    scores[o] = masked;
    clsid[o]  = bestc;
}

// ---------------------------------------------------------------------------
// Kernel 2: per-batch full descending sort + gather, entirely in LDS.
//   LDS layout (299,072 B < 320 KB, CDNA5-only):
//     [0,        131072)  keys   : 16384 x u64 bitonic sort array
//     [131072,   265472)  boxes  : 8400 x float4 tile (TDM #3, overlapped)
//                                  -- front 33.6 KB aliases the score stage
//     [265472,   299072)  cls    : 8400 x i32 tile (TDM #2)
//   TDM #1/#2 (scores, cls) complete before key packing; TDM #3 (boxes) runs
//   asynchronously UNDER the 105-pass bitonic sort; s_wait_tensorcnt at end.
// ---------------------------------------------------------------------------
#define OFF_KEYS  0
#define OFF_BOXES 131072
#define OFF_CLS   265472

__global__ void __launch_bounds__(1024)
sort_kernel(const float* __restrict__ boxes,
            const float* __restrict__ scores,
            const int*   __restrict__ clsid,
            float* __restrict__ out_boxes,   // [B*NPB*4]
            float* __restrict__ out_scores,  // [B*NPB]
            int*   __restrict__ out_cls,     // [B*NPB]
            int*   __restrict__ out_count)   // [B]
{
    extern __shared__ unsigned char smem[];
    uint64_t* keys     = reinterpret_cast<uint64_t*>(smem + OFF_KEYS);
    float*    stage    = reinterpret_cast<float*>(smem + OFF_BOXES); // aliased
    float4*   boxesLds = reinterpret_cast<float4*>(smem + OFF_BOXES);
    int*      clsLds   = reinterpret_cast<int*>(smem + OFF_CLS);
    __shared__ int vcount;

    const int b   = blockIdx.x;
    const int tid = threadIdx.x;
    const int NT  = 1024;
    const size_t ob = (size_t)b * NPB;

    // --- TDM #1 + #2: scores and class ids, global -> LDS -------------------
#ifdef HAVE_TDM
    if (tid < 32) {   // wave 0 issues; TENSORcnt is per-wave
        tdm_copy_dwords((uint32_t)(uintptr_t)(void*)stage,  scores + ob, NPB);
        tdm_copy_dwords((uint32_t)(uintptr_t)(void*)clsLds, clsid  + ob, NPB);
        __builtin_amdgcn_s_wait_tensorcnt(0);
    }
    __syncthreads();   // other 31 waves wait for wave 0's DMAs
#else
    for (int i = tid; i < NPB; i += NT) {
        stage[i]  = scores[ob + i];
        clsLds[i] = clsid[ob + i];
    }
    __syncthreads();
#endif

    // Pack sortable keys; pad to 16384 with 0 (sorts after -1.0 sentinels).
    for (int i = tid; i < NSORT; i += NT) {
        uint64_t k = 0;
        if (i < NPB) {
            uint32_t bits = __float_as_uint(stage[i]);
            uint32_t sb = (bits & 0x80000000u) ? ~bits : (bits | 0x80000000u);
            k = ((uint64_t)sb << 32) | (uint32_t)(~(uint32_t)i);
        }
        keys[i] = k;
    }
    if (tid == 0) vcount = 0;
    __syncthreads();   // all stage readers done -> safe to overwrite region

    // --- TDM #3: box tile DMA runs concurrently with the whole sort ---------
#ifdef HAVE_TDM
    if (tid < 32) {
        tdm_copy_dwords((uint32_t)(uintptr_t)(void*)boxesLds,
                        boxes + ob * 4, NPB * 4);      // 33600 dwords, async
    }
#else
    for (int i = tid; i < NPB; i += NT)
        boxesLds[i] = reinterpret_cast<const float4*>(boxes)[ob + i];
#endif

    // Bitonic sort, descending.  105 compare-exchange passes over LDS keys.
    for (int k = 2; k <= NSORT; k <<= 1) {
        for (int j = k >> 1; j > 0; j >>= 1) {
            for (int i = tid; i < NSORT; i += NT) {
                int ij = i ^ j;
                if (ij > i) {
                    uint64_t a0 = keys[i], a1 = keys[ij];
                    bool desc = ((i & k) == 0);
                    if (desc ? (a0 < a1) : (a0 > a1)) {
                        keys[i] = a1; keys[ij] = a0;
                    }
                }
            }
            __syncthreads();
        }
    }

    // Box DMA finished long ago; retire it and make LDS visible to all waves.
#ifdef HAVE_TDM
    if (tid < 32) __builtin_amdgcn_s_wait_tensorcnt(0);
#endif
    __syncthreads();

    // Emit: filtered scores + LDS-gathered classes/boxes + valid count.
    int local = 0;
    for (int r = tid; r < NPB; r += NT) {
        uint64_t kk = keys[r];
        uint32_t sb = (uint32_t)(kk >> 32);
        uint32_t bits = (sb & 0x80000000u) ? (sb & 0x7FFFFFFFu) : ~sb;
        float s = __uint_as_float(bits);
        int idx = (int)(~(uint32_t)(kk & 0xFFFFFFFFu));
        bool valid = s > 0.0f;
        local += valid ? 1 : 0;
        out_scores[ob + r] = valid ? s : 0.0f;
        out_cls[ob + r]    = clsLds[idx];
        reinterpret_cast<float4*>(out_boxes)[ob + r] = boxesLds[idx];
    }
    atomicAdd(&vcount, local);
    __syncthreads();
    if (tid == 0) out_count[b] = vcount;
}

// ---------------------------------------------------------------------------
extern "C" void kernel_launch(void* const* d_in, const int* in_sizes, int n_in,
                              void* d_out, int out_size, void* d_ws, size_t ws_size,
                              hipStream_t stream) {
    const float* out0 = (const float*)d_in[0];
    const float* out1 = (const float*)d_in[1];
    const float* out2 = (const float*)d_in[2];

    // Workspace: boxes | masked scores | class ids  (~6.5 MB, L2-resident)
    float* ws_boxes  = (float*)d_ws;
    float* ws_scores = ws_boxes + (size_t)BATCH * NPB * 4;
    int*   ws_cls    = (int*)(ws_scores + (size_t)BATCH * NPB);

    // Outputs concatenated flat in reference return order.
    float* o_boxes  = (float*)d_out;
    float* o_scores = o_boxes + (size_t)BATCH * NPB * 4;
    int*   o_cls    = (int*)(o_scores + (size_t)BATCH * NPB);
    int*   o_count  = o_cls + (size_t)BATCH * NPB;

    int total = BATCH * NPB;
    decode_kernel<<<(total + 255) / 256, 256, 0, stream>>>(
        out0, out1, out2, ws_boxes, ws_scores, ws_cls);

    size_t shmem = OFF_CLS + (size_t)NPB * 4;   // 299,072 B < 320 KB WGP LDS
    sort_kernel<<<BATCH, 1024, shmem, stream>>>(
        ws_boxes, ws_scores, ws_cls, o_boxes, o_scores, o_cls, o_count);
}